// ENNBICEPHybrid_47502338294457
// MI455X (gfx1250) — compile-verified
//
#include <hip/hip_runtime.h>

typedef __attribute__((ext_vector_type(2))) float v2f;
typedef __attribute__((ext_vector_type(4))) float v4f;
typedef __attribute__((ext_vector_type(8))) float v8f;

#define B_   512
#define IN_  512
#define T_   1024
#define P_   128
#define OUT_ 1024

// workspace layout in floats
#define WS_FBC   0
#define WS_G     512
#define WS_H     1536
#define WS_CB    2560
#define WS_U     3584                    // [512][384]  (S | last | first), pre-scaled by c_b
#define WS_CWT   (3584 + 512 * 384)      // [1024][384] fused weight, transposed [j][k]

// ---------------- per-batch feedback scalar: fbc[b] = sigmoid(x[b]·W_fb + b_fb) * sqrt(dt)
__global__ void fb_kernel(const float* __restrict__ x, const float* __restrict__ Wfb,
                          const float* __restrict__ bfb, float* __restrict__ fbc) {
  int b = blockIdx.x;
  int lane = threadIdx.x;          // 32 threads = 1 wave
  const float* xr = x + b * IN_;
  float acc = 0.f;
  for (int i = lane; i < IN_; i += 32) acc += xr[i] * Wfb[i];
  for (int off = 16; off > 0; off >>= 1) acc += __shfl_xor(acc, off, 32);
  if (lane == 0) {
    float z = acc + bfb[0];
    float fb = 1.f / (1.f + expf(-z));
    fbc[b] = fb * 0.03125f;        // sqrt(1/1024) = 1/32
  }
}

// ---------------- geometric-series time weights g[t], h[t]
__global__ void weights_kernel(const float* __restrict__ dc,
                               float* __restrict__ g, float* __restrict__ h) {
  int t = blockIdx.x * blockDim.x + threadIdx.x;
  if (t >= T_) return;
  double decay = 0.5 / (1.0 + exp(-(double)dc[0]));
  double a = 1.0 - decay / (double)T_;
  double hp = pow(a, (double)(T_ - 1 - t));      // a^(1023-t)
  h[t] = (float)hp;
  g[t] = (float)((1.0 - hp * a) / (1.0 - a));    // (1 - a^(1024-t)) / (1-a)
}

// ---------------- fused weight matrix CwT[j][k] = sum_o M1[k][o] * W_out[j][o], k = (kk*128 + i)
// kk=0 -> (w0+w1+w2)/T  (multiplies S), kk=1 -> -w0/T (multiplies last), kk=2 -> -w2/T (multiplies first)
__global__ void cwt_kernel(const float* __restrict__ conv_w, const float* __restrict__ W_out,
                           float* __restrict__ CwT) {
  int idx = blockIdx.x * blockDim.x + threadIdx.x;   // < 1024*384
  int j = idx / 384;
  int k = idx - j * 384;
  int kk = k >> 7;
  int i = k & 127;
  const float* wrow = W_out + (j << 7);
  float acc = 0.f;
  for (int o = 0; o < P_; ++o) {
    const float* w3 = conv_w + (size_t)((o << 7) + i) * 3;
    float m1 = (kk == 0) ? (w3[0] + w3[1] + w3[2]) : ((kk == 1) ? -w3[0] : -w3[2]);
    acc += m1 * wrow[o];
  }
  CwT[idx] = acc * (1.0f / (float)T_);
}

// ---------------- fused bias cb[j] = conv_b · W_out[j] + b_out[j]
__global__ void cb_kernel(const float* __restrict__ conv_b, const float* __restrict__ W_out,
                          const float* __restrict__ b_out, float* __restrict__ cb) {
  int j = blockIdx.x * blockDim.x + threadIdx.x;
  if (j >= OUT_) return;
  const float* wrow = W_out + (j << 7);
  float acc = b_out[j];
  for (int o = 0; o < P_; ++o) acc += conv_b[o] * wrow[o];
  cb[j] = acc;
}

// ---------------- bandwidth-bound pass over noise: U[b] = c_b * [Σ g_t z | Σ h_t z | z_0]
__global__ void reduce_kernel(const float* __restrict__ noise, const float* __restrict__ fbc,
                              const float* __restrict__ g, const float* __restrict__ h,
                              float* __restrict__ U) {
  int b = blockIdx.x;
  int lane = threadIdx.x & 31;
  int wv = threadIdx.x >> 5;          // 8 waves split T into 8 chunks of 128
  const float* base = noise + (size_t)b * (T_ * P_) + (size_t)wv * 128 * P_ + lane * 4;
  v4f s = {0.f, 0.f, 0.f, 0.f};
  v4f l = {0.f, 0.f, 0.f, 0.f};
  #pragma unroll 4
  for (int tt = 0; tt < 128; ++tt) {
    int t = wv * 128 + tt;
    v4f z = *(const v4f*)(base + (size_t)tt * P_);   // coalesced b128: wave covers all 128 p
    float gg = g[t];
    float hh = h[t];
    s += gg * z;
    l += hh * z;
  }
  __shared__ v4f shS[8][32];
  __shared__ v4f shL[8][32];
  shS[wv][lane] = s;
  shL[wv][lane] = l;
  __syncthreads();
  if (wv == 0) {
    v4f S = shS[0][lane];
    v4f L = shL[0][lane];
    #pragma unroll
    for (int w = 1; w < 8; ++w) { S += shS[w][lane]; L += shL[w][lane]; }   // fixed order: deterministic
    v4f F = *(const v4f*)(noise + (size_t)b * (T_ * P_) + lane * 4);        // t = 0 row
    float c = fbc[b];
    float* urow = U + b * 384;
    *(v4f*)(urow + lane * 4)       = c * S;
    *(v4f*)(urow + 128 + lane * 4) = c * L;
    *(v4f*)(urow + 256 + lane * 4) = c * F;
  }
}

// ---------------- out[512,1024] = U[512,384] @ CwT^T + cb   via V_WMMA_F32_16X16X4_F32
// One 16x16 tile per wave. A-fragment (16x4 f32): lane l, vgpr v -> A[m=l%16, k=v+(l/16)*2]
// B-fragment (4x16 f32):  lane l, vgpr v -> B[k=v+(l/16)*2, n=l%16] = CwT[n][k]  -> both are float2 loads.
__global__ void gemm_wmma_kernel(const float* __restrict__ U, const float* __restrict__ CwT,
                                 const float* __restrict__ cb, float* __restrict__ out) {
  int lane = threadIdx.x & 31;
  int wv = threadIdx.x >> 5;                 // 8 waves -> 8 N-tiles per block
  int m0 = blockIdx.y * 16;
  int n0 = (blockIdx.x * 8 + wv) * 16;
  int r = lane & 15;                         // m for A, n for B/C/D
  int koff = (lane >> 4) << 1;               // 0 or 2
  const float* Ap = U + (size_t)(m0 + r) * 384 + koff;
  const float* Bp = CwT + (size_t)(n0 + r) * 384 + koff;
  v8f c = {};
  #pragma unroll 8
  for (int k = 0; k < 384; k += 4) {
    v2f a = *(const v2f*)(Ap + k);
    v2f bfrag = *(const v2f*)(Bp + k);
    c = __builtin_amdgcn_wmma_f32_16x16x4_f32(false, a, false, bfrag,
                                              (short)0, c, false, false);
  }
  float bias = cb[n0 + r];
  int mhalf = (lane >> 4) << 3;              // lanes 16-31 hold rows m0+8..m0+15
  #pragma unroll
  for (int v = 0; v < 8; ++v) {
    out[(size_t)(m0 + mhalf + v) * OUT_ + (n0 + r)] = c[v] + bias;
  }
}

extern "C" void kernel_launch(void* const* d_in, const int* in_sizes, int n_in,
                              void* d_out, int out_size, void* d_ws, size_t ws_size,
                              hipStream_t stream) {
  const float* x      = (const float*)d_in[0];
  const float* noise  = (const float*)d_in[1];
  const float* W_fb   = (const float*)d_in[2];
  const float* b_fb   = (const float*)d_in[3];
  const float* dc     = (const float*)d_in[4];
  const float* conv_w = (const float*)d_in[5];
  const float* conv_b = (const float*)d_in[6];
  const float* W_out  = (const float*)d_in[7];
  const float* b_out  = (const float*)d_in[8];
  float* out = (float*)d_out;

  float* ws  = (float*)d_ws;
  float* fbc = ws + WS_FBC;
  float* g   = ws + WS_G;
  float* h   = ws + WS_H;
  float* cb  = ws + WS_CB;
  float* U   = ws + WS_U;
  float* CwT = ws + WS_CWT;

  fb_kernel<<<B_, 32, 0, stream>>>(x, W_fb, b_fb, fbc);
  weights_kernel<<<T_ / 256, 256, 0, stream>>>(dc, g, h);
  cwt_kernel<<<(OUT_ * 384) / 256, 256, 0, stream>>>(conv_w, W_out, CwT);
  cb_kernel<<<OUT_ / 256, 256, 0, stream>>>(conv_b, W_out, b_out, cb);
  reduce_kernel<<<B_, 256, 0, stream>>>(noise, fbc, g, h, U);
  dim3 grid(OUT_ / (16 * 8), B_ / 16);       // (8, 32), 8 waves/block
  gemm_wmma_kernel<<<grid, 256, 0, stream>>>(U, CwT, cb, out);
}